// BBConv_31061203485066
// MI455X (gfx1250) — compile-verified
//
#include <hip/hip_runtime.h>
#include <math.h>

typedef __attribute__((ext_vector_type(2))) float v2f;
typedef __attribute__((ext_vector_type(8))) float v8f;

#define DIN 128
#define DOUT 128
#define GEMM_ROWS 64   // rows per block: 4 row-tiles of 16

// ---------------------------------------------------------------------------
// K0: X = F @ W, fp32 via V_WMMA_F32_16X16X4_F32.
// Block = 256 threads = 8 waves. Block covers 64 rows x 128 cols.
// Each wave owns a 16-wide column strip and keeps 4 accumulators (4 row-tiles),
// reusing every B-fragment 4x to amortize the 64KB weight load in LDS.
// ---------------------------------------------------------------------------
__global__ __launch_bounds__(256) void gcn_gemm_wmma(
    const float* __restrict__ F, const float* __restrict__ W,
    float* __restrict__ X, int N)
{
    __shared__ float sW[DIN * DOUT];        // 64 KB
    __shared__ float sF[GEMM_ROWS * DIN];   // 32 KB
    const int tid  = threadIdx.x;
    const int row0 = blockIdx.x * GEMM_ROWS;

    // Cooperative load of the full 128x128 weight (float4, coalesced).
    const float4* W4  = reinterpret_cast<const float4*>(W);
    float4*       sW4 = reinterpret_cast<float4*>(sW);
#pragma unroll
    for (int i = 0; i < (DIN * DOUT / 4) / 256; ++i)
        sW4[tid + i * 256] = W4[tid + i * 256];

    // Cooperative load of the 64xDIN feature tile (float4, row-guarded).
    const float4* F4  = reinterpret_cast<const float4*>(F);
    float4*       sF4 = reinterpret_cast<float4*>(sF);
#pragma unroll
    for (int i = 0; i < (GEMM_ROWS * DIN / 4) / 256; ++i) {
        int j = tid + i * 256;            // float4 index within tile
        int r = row0 + (j >> 5);          // DIN/4 = 32 float4 per row
        float4 z; z.x = z.y = z.z = z.w = 0.0f;
        sF4[j] = (r < N) ? F4[(size_t)r * (DIN / 4) + (j & 31)] : z;
    }
    __syncthreads();

    const int wave = tid >> 5;
    const int lane = tid & 31;
    const int m    = lane & 15;   // row within tile (A) / col within tile (B,D)
    const int half = lane >> 4;   // selects K pair (A,B) / M half (D)
    const int n    = wave * 16 + m;

    v8f acc[4] = {};
#pragma unroll
    for (int k0 = 0; k0 < DIN; k0 += 4) {
        const int kb = k0 + half * 2;
        // B 4x16 fragment: lane L -> B[kb][n], B[kb+1][n]  (shared by 4 tiles)
        v2f b;
        b.x = sW[kb * DOUT + n];
        b.y = sW[(kb + 1) * DOUT + n];
#pragma unroll
        for (int t = 0; t < 4; ++t) {
            // A 16x4 fragment: lane L -> A[t*16 + (L&15)][kb .. kb+1]
            const v2f a = *reinterpret_cast<const v2f*>(&sF[(t * 16 + m) * DIN + kb]);
            acc[t] = __builtin_amdgcn_wmma_f32_16x16x4_f32(
                /*neg_a=*/false, a, /*neg_b=*/false, b,
                /*c_mod=*/(short)0, acc[t], /*reuse_a=*/false, /*reuse_b=*/false);
        }
    }

    // D layout: VGPR i -> row (i + 8*half), col (lane&15)
#pragma unroll
    for (int t = 0; t < 4; ++t) {
#pragma unroll
        for (int i = 0; i < 8; ++i) {
            int r = row0 + t * 16 + half * 8 + i;
            if (r < N) X[(size_t)r * DOUT + n] = acc[t][i];
        }
    }
}

// ---------------------------------------------------------------------------
// CSR build (unsorted segments via global cursor — only integer atomics).
// ---------------------------------------------------------------------------
__global__ void gcn_init_counts(int* counts, int* total, int N)
{
    int i = blockIdx.x * blockDim.x + threadIdx.x;
    if (i < N) counts[i] = 0;
    if (i == 0) *total = 0;
}

__global__ void gcn_count_edges(const int* __restrict__ rowidx, int* counts, int E)
{
    int e = blockIdx.x * blockDim.x + threadIdx.x;
    if (e < E) atomicAdd(&counts[rowidx[e]], 1);
}

__global__ void gcn_alloc_rows(const int* __restrict__ counts,
                               int* rstart, int* cursor, int* total, int N)
{
    int r = blockIdx.x * blockDim.x + threadIdx.x;
    if (r < N) {
        int c = counts[r];
        int s = atomicAdd(total, c);
        rstart[r] = s;
        cursor[r] = s;
    }
}

__global__ void gcn_scatter_edges(const int* __restrict__ idx,
                                  const float* __restrict__ vals,
                                  int* cursor, int* __restrict__ ccol,
                                  float* __restrict__ cval, int E)
{
    int e = blockIdx.x * blockDim.x + threadIdx.x;
    if (e < E) {
        int r = idx[e];            // row
        int p = atomicAdd(&cursor[r], 1);
        ccol[p] = idx[E + e];      // col
        cval[p] = vals[e];
    }
}

// ---------------------------------------------------------------------------
// K5: fused SpMM + bias + ELU + LayerNorm. One wave32 per output row.
// Each lane holds 4 features; edges broadcast via __shfl. The 32-edge full
// chunk is fully unrolled so loads queue up before the first wait.
// ---------------------------------------------------------------------------
__global__ __launch_bounds__(256) void gcn_spmm_elu_ln(
    const float* __restrict__ X, const int* __restrict__ rstart,
    const int* __restrict__ rcnt, const int* __restrict__ ccol,
    const float* __restrict__ cval, const float* __restrict__ bias,
    const float* __restrict__ gamma, const float* __restrict__ beta,
    float* __restrict__ out, int N)
{
    const int row  = (int)((blockIdx.x * blockDim.x + threadIdx.x) >> 5);
    const int lane = threadIdx.x & 31;
    if (row >= N) return;

    const int s = rstart[row];
    const int e = s + rcnt[row];
    const float* __restrict__ Xl = X + lane * 4;   // per-lane column base

    float acc0 = 0.f, acc1 = 0.f, acc2 = 0.f, acc3 = 0.f;
    for (int base = s; base < e; base += 32) {
        const int idx  = base + lane;
        const int left = e - base;
        int   c = 0;
        float v = 0.0f;
        if (idx < e) { c = ccol[idx]; v = cval[idx]; }

        if (left >= 32) {
            // Fast path: full 32-edge chunk, fully unrolled for load ILP.
#pragma unroll
            for (int j = 0; j < 32; ++j) {
                const int   cj = __shfl(c, j, 32);
                const float vj = __shfl(v, j, 32);
                const float4 xr =
                    *reinterpret_cast<const float4*>(Xl + (size_t)cj * DOUT);
                acc0 = fmaf(vj, xr.x, acc0);
                acc1 = fmaf(vj, xr.y, acc1);
                acc2 = fmaf(vj, xr.z, acc2);
                acc3 = fmaf(vj, xr.w, acc3);
            }
        } else {
            for (int j = 0; j < left; ++j) {
                const int   cj = __shfl(c, j, 32);
                const float vj = __shfl(v, j, 32);
                const float4 xr =
                    *reinterpret_cast<const float4*>(Xl + (size_t)cj * DOUT);
                acc0 = fmaf(vj, xr.x, acc0);
                acc1 = fmaf(vj, xr.y, acc1);
                acc2 = fmaf(vj, xr.z, acc2);
                acc3 = fmaf(vj, xr.w, acc3);
            }
        }
    }

    // bias + ELU
    const float4 b4 = *reinterpret_cast<const float4*>(bias + lane * 4);
    float h0 = acc0 + b4.x, h1 = acc1 + b4.y, h2 = acc2 + b4.z, h3 = acc3 + b4.w;
    h0 = (h0 > 0.f) ? h0 : expm1f(h0);
    h1 = (h1 > 0.f) ? h1 : expm1f(h1);
    h2 = (h2 > 0.f) ? h2 : expm1f(h2);
    h3 = (h3 > 0.f) ? h3 : expm1f(h3);

    // mean over 128 features (wave32 xor-reduction)
    float sum = h0 + h1 + h2 + h3;
#pragma unroll
    for (int o = 16; o >= 1; o >>= 1) sum += __shfl_xor(sum, o, 32);
    const float mu = sum * (1.0f / DOUT);

    float d0 = h0 - mu, d1 = h1 - mu, d2 = h2 - mu, d3 = h3 - mu;
    float vs = d0 * d0 + d1 * d1 + d2 * d2 + d3 * d3;
#pragma unroll
    for (int o = 16; o >= 1; o >>= 1) vs += __shfl_xor(vs, o, 32);
    const float rstd = rsqrtf(vs * (1.0f / DOUT) + 1e-5f);

    const float4 g4  = *reinterpret_cast<const float4*>(gamma + lane * 4);
    const float4 be4 = *reinterpret_cast<const float4*>(beta + lane * 4);
    float4 o4;
    o4.x = d0 * rstd * g4.x + be4.x;
    o4.y = d1 * rstd * g4.y + be4.y;
    o4.z = d2 * rstd * g4.z + be4.z;
    o4.w = d3 * rstd * g4.w + be4.w;
    *reinterpret_cast<float4*>(out + (size_t)row * DOUT + lane * 4) = o4;
}

// ---------------------------------------------------------------------------
extern "C" void kernel_launch(void* const* d_in, const int* in_sizes, int n_in,
                              void* d_out, int out_size, void* d_ws, size_t ws_size,
                              hipStream_t stream)
{
    const int*   indices  = (const int*)d_in[0];   // [2, E] (row ; col)
    const float* values   = (const float*)d_in[1]; // [E]
    const float* features = (const float*)d_in[2]; // [N, 128]
    const float* weight   = (const float*)d_in[3]; // [128, 128]
    const float* bias     = (const float*)d_in[4]; // [128]
    const float* gamma    = (const float*)d_in[5]; // [128]
    const float* beta     = (const float*)d_in[6]; // [128]
    float*       out      = (float*)d_out;

    const int E = in_sizes[1];
    const int N = in_sizes[2] / DIN;

    // Workspace carve-up (256B aligned slices)
    char*  ws  = (char*)d_ws;
    size_t off = 0;
    auto carve = [&](size_t bytes) -> void* {
        void* p = ws + off;
        off = (off + bytes + 255) & ~(size_t)255;
        return p;
    };
    float* X      = (float*)carve((size_t)N * DOUT * sizeof(float));
    int*   counts = (int*)carve((size_t)N * sizeof(int));
    int*   rstart = (int*)carve((size_t)N * sizeof(int));
    int*   cursor = (int*)carve((size_t)N * sizeof(int));
    int*   total  = (int*)carve(256);
    int*   ccol   = (int*)carve((size_t)E * sizeof(int));
    float* cval   = (float*)carve((size_t)E * sizeof(float));
    (void)ws_size; (void)n_in; (void)out_size;

    const int nBlkGemm = (N + GEMM_ROWS - 1) / GEMM_ROWS;
    const int nBlkN    = (N + 255) / 256;
    const int nBlkE    = (E + 255) / 256;
    const int nBlkRow  = (N + 7) / 8;   // 8 rows (waves) per 256-thread block

    gcn_gemm_wmma<<<nBlkGemm, 256, 0, stream>>>(features, weight, X, N);
    gcn_init_counts<<<nBlkN, 256, 0, stream>>>(counts, total, N);
    gcn_count_edges<<<nBlkE, 256, 0, stream>>>(indices, counts, E);
    gcn_alloc_rows<<<nBlkN, 256, 0, stream>>>(counts, rstart, cursor, total, N);
    gcn_scatter_edges<<<nBlkE, 256, 0, stream>>>(indices, values, cursor, ccol, cval, E);
    gcn_spmm_elu_ln<<<nBlkRow, 256, 0, stream>>>(X, rstart, counts, ccol, cval,
                                                 bias, gamma, beta, out, N);
}